// StatefulLlamaDecodeStepBeforeNorm_20160576488109
// MI455X (gfx1250) — compile-verified
//
#include <hip/hip_runtime.h>
#include <hip/hip_bf16.h>
#include <math.h>

// ---------------- model constants ----------------
#define LAYERS 4
#define DMODEL 2048
#define NHEADS 16
#define NKVH   8
#define HDIM   128
#define DFF    5632
#define SEQ    8192
#define NSPLIT 32
#define ATT_SCALE 0.08838834764831845f   // 128^-0.5

typedef float v2f __attribute__((ext_vector_type(2)));
typedef float v8f __attribute__((ext_vector_type(8)));

// B-operand preprocessing modes for the fused GEMV
#define B_PLAIN  0   // b[k] = xa[k]
#define B_RMS    1   // b[k] = xa[k] * xb[k] * rsqrt(mean(xa^2)+eps)   (RMSNorm fused)
#define B_SWIGLU 2   // b[k] = silu(xa[k]) * xb[k]                     (SwiGLU fused)

// ---------------------------------------------------------------------------
// WMMA f32 16x16x4 rank-4 accumulate:  D[i][j] += sum_k A[i][k]*B[k][j]
// A layout (ISA 7.12.2): lane L<16 holds A[L][0],A[L][1] in {v0,v1};
//                        lane L>=16 holds A[L-16][2],A[L-16][3].
// B is K-striped the same way; we broadcast x across all 16 columns, so every
// column of D holds the same partial y for 16 output rows.
// ---------------------------------------------------------------------------
__device__ __forceinline__ v8f wmma4(v2f a, v2f b, v8f c) {
    return __builtin_amdgcn_wmma_f32_16x16x4_f32(
        /*neg_a=*/false, a, /*neg_b=*/false, b,
        /*c_mod=*/(short)0, c, /*reuse_a=*/false, /*reuse_b=*/false);
}

// ---------------------------------------------------------------------------
// Fused GEMV via WMMA over up to 3 row-concatenated weight matrices:
//   y[m] = (res ? res[m] : 0) + sum_k W(m,k) * b[k]
// Block = 256 threads = 8 waves = 2 tiles x 4 k-splits; grid = totalTiles/2.
// tile < b0 -> W0 ; tile < b1 -> W1 ; else W2.  Output rows are contiguous.
// The shared b[k] vector is materialized once per block in LDS; the hot loop
// is 4x global_load_b64 (W) + 4x ds_load_b64 (b) + 4x v_wmma per 16 K.
// Requires (16*tiles) % 32 == 0 and K % 64 == 0, K <= DFF.
// ---------------------------------------------------------------------------
template <int BMODE>
__global__ __launch_bounds__(256)
void gemv_fused_kernel(const float* __restrict__ W0, const float* __restrict__ W1,
                       const float* __restrict__ W2, int b0, int b1,
                       const float* __restrict__ xa, const float* __restrict__ xb,
                       const float* __restrict__ res, float* __restrict__ y, int K) {
    __shared__ float bstage[DFF];        // staged B vector (first K entries used)
    __shared__ float red[2][4][16];
    __shared__ float sred[256];

    const int tid  = threadIdx.x;
    const int wave = tid >> 5;
    const int lane = tid & 31;
    const int half = lane >> 4;      // 0: K+{0,1}, 1: K+{2,3}
    const int r    = lane & 15;      // row within tile
    const int tloc = wave >> 2;      // tile within block (0..1)
    const int ks   = wave & 3;       // k-split (0..3)
    const int tile = blockIdx.x * 2 + tloc;

    // ---- stage b[k] into LDS (all 256 threads cooperate) ----
    if (BMODE == B_RMS) {
        float acc = 0.0f;
        for (int i = tid; i < K; i += 256) { float v = xa[i]; acc += v * v; }
        sred[tid] = acc;
        __syncthreads();
        for (int st = 128; st > 0; st >>= 1) {
            if (tid < st) sred[tid] += sred[tid + st];
            __syncthreads();
        }
        const float inv = rsqrtf(sred[0] * (1.0f / (float)K) + 1e-6f);
        for (int i = tid; i < K; i += 256) bstage[i] = xa[i] * xb[i] * inv;
    } else if (BMODE == B_SWIGLU) {
        for (int i = tid; i < K; i += 256) {
            const float g = xa[i];
            bstage[i] = (g / (1.0f + __expf(-g))) * xb[i];
        }
    } else {
        for (int i = tid; i < K; i += 256) bstage[i] = xa[i];
    }
    __syncthreads();

    // ---- select source matrix for this tile ----
    const float* Wsel;
    int trel;
    if (tile < b0)      { Wsel = W0; trel = tile; }
    else if (tile < b1) { Wsel = W1; trel = tile - b0; }
    else                { Wsel = W2; trel = tile - b1; }

    const int row = trel * 16 + r;
    const int kq  = K >> 2;          // per-k-split range
    const int k0b = ks * kq;
    const int off = half * 2;

    const float* __restrict__ wp = Wsel + (size_t)row * (size_t)K + k0b + off;
    const float* bp = bstage + k0b + off;          // LDS pointer (broadcast reads)
    const int niter = kq >> 4;                     // 16 K per iteration

    v8f c0 = {}; v8f c1 = {}; v8f c2 = {}; v8f c3 = {};
    for (int it = 0; it < niter; ++it) {
        __builtin_prefetch(wp + 1024, 0, 0);       // ~4KB ahead per row
        v2f a0 = *(const v2f*)(wp +  0);
        v2f a1 = *(const v2f*)(wp +  4);
        v2f a2 = *(const v2f*)(wp +  8);
        v2f a3 = *(const v2f*)(wp + 12);
        v2f f0 = *(const v2f*)(bp +  0);
        v2f f1 = *(const v2f*)(bp +  4);
        v2f f2 = *(const v2f*)(bp +  8);
        v2f f3 = *(const v2f*)(bp + 12);
        c0 = wmma4(a0, f0, c0);
        c1 = wmma4(a1, f1, c1);
        c2 = wmma4(a2, f2, c2);
        c3 = wmma4(a3, f3, c3);
        wp += 16;
        bp += 16;
    }
    v8f c = c0 + c1 + c2 + c3;       // columns identical -> partial y per M-row

    // D layout: lane0 vgpr i -> M=i (i=0..7); lane16 vgpr i -> M=8+i.
    if (r == 0) {
        float* dst = &red[tloc][ks][half * 8];
        #pragma unroll
        for (int i = 0; i < 8; ++i) dst[i] = c[i];
    }
    __syncthreads();

    if (tid < 32) {
        const int t = tid >> 4, m = tid & 15;
        const float s = red[t][0][m] + red[t][1][m] + red[t][2][m] + red[t][3][m];
        const int orow = (blockIdx.x * 2 + t) * 16 + m;
        y[orow] = (res ? res[orow] : 0.0f) + s;
    }
}

// ---------------------------------------------------------------------------
__global__ void embed_kernel(const float* __restrict__ embed_w,
                             const int* __restrict__ ids, float* __restrict__ x) {
    const int tok = ids[0];
    const float* src = embed_w + (size_t)tok * DMODEL;
    for (int i = threadIdx.x; i < DMODEL; i += blockDim.x) x[i] = src[i];
}

// ---------------------------------------------------------------------------
// Flash-decoding phase 1 with fused RoPE: grid (NHEADS, NSPLIT), 8 waves.
// Lane L holds dims 4L..4L+3; the rotate-half partner (d +/- 64) lives exactly
// 16 lanes away -> exchange via __shfl_xor(.,16).
// ---------------------------------------------------------------------------
__global__ __launch_bounds__(256)
void attn_partial_kernel(const float* __restrict__ q,  const float* __restrict__ kn,
                         const float* __restrict__ vn, const float* __restrict__ kc,
                         const float* __restrict__ vc, const int* __restrict__ cpos,
                         float* __restrict__ pm, float* __restrict__ pl,
                         float* __restrict__ po) {
    const int h     = blockIdx.x;
    const int split = blockIdx.y;
    const int pos   = cpos[0];
    const int total = pos + 1;
    const int chunk = (total + NSPLIT - 1) / NSPLIT;
    const int s0    = split * chunk;
    const int s1    = (s0 + chunk < total) ? (s0 + chunk) : total;

    const int lane = threadIdx.x & 31;
    const int wave = threadIdx.x >> 5;
    const int kh   = h >> 1;                         // N_REP = 2
    const float pf = (float)pos;

    // per-lane RoPE coefficients for dims d = 4*lane + c
    float cc[4], sc[4];
    const float sign = (lane < 16) ? -1.0f : 1.0f;   // d<64 : -rot ; d>=64 : +rot
    #pragma unroll
    for (int cmp = 0; cmp < 4; ++cmp) {
        const int j = (lane * 4 + cmp) & 63;
        const float ang = pf * __powf(10000.0f, -(float)(2 * j) * (1.0f / (float)HDIM));
        cc[cmp] = cosf(ang);
        sc[cmp] = sinf(ang);
    }

    const float* __restrict__ kbase = kc + (size_t)kh * SEQ * HDIM;
    const float* __restrict__ vbase = vc + (size_t)kh * SEQ * HDIM;

    // load + rope q fragment
    float4 qf = *(const float4*)(q + h * HDIM + lane * 4);
    {
        float4 qp;
        qp.x = __shfl_xor(qf.x, 16, 32); qp.y = __shfl_xor(qf.y, 16, 32);
        qp.z = __shfl_xor(qf.z, 16, 32); qp.w = __shfl_xor(qf.w, 16, 32);
        qf.x = qf.x * cc[0] + sign * qp.x * sc[0];
        qf.y = qf.y * cc[1] + sign * qp.y * sc[1];
        qf.z = qf.z * cc[2] + sign * qp.z * sc[2];
        qf.w = qf.w * cc[3] + sign * qp.w * sc[3];
    }

    float m = -INFINITY, l = 0.0f;
    float4 acc = {0.f, 0.f, 0.f, 0.f};

    for (int s = s0 + wave; s < s1; s += 8) {
        float4 kf;
        if (s == pos) {
            kf = *(const float4*)(kn + kh * HDIM + lane * 4);   // rope the new k
            float4 kp;
            kp.x = __shfl_xor(kf.x, 16, 32); kp.y = __shfl_xor(kf.y, 16, 32);
            kp.z = __shfl_xor(kf.z, 16, 32); kp.w = __shfl_xor(kf.w, 16, 32);
            kf.x = kf.x * cc[0] + sign * kp.x * sc[0];
            kf.y = kf.y * cc[1] + sign * kp.y * sc[1];
            kf.z = kf.z * cc[2] + sign * kp.z * sc[2];
            kf.w = kf.w * cc[3] + sign * kp.w * sc[3];
        } else {
            kf = *(const float4*)(kbase + (size_t)s * HDIM + lane * 4);
        }
        float d = qf.x * kf.x + qf.y * kf.y + qf.z * kf.z + qf.w * kf.w;
        #pragma unroll
        for (int o = 16; o > 0; o >>= 1) d += __shfl_xor(d, o, 32);
        const float score = d * ATT_SCALE;
        const float mn   = fmaxf(m, score);
        const float corr = __expf(m - mn);
        const float p    = __expf(score - mn);
        const float* vr = (s == pos) ? (vn + kh * HDIM) : (vbase + (size_t)s * HDIM);
        const float4 vf = *(const float4*)(vr + lane * 4);
        acc.x = acc.x * corr + p * vf.x;
        acc.y = acc.y * corr + p * vf.y;
        acc.z = acc.z * corr + p * vf.z;
        acc.w = acc.w * corr + p * vf.w;
        l = l * corr + p;
        m = mn;
    }

    __shared__ float sm[8], sl[8];
    __shared__ float so[8][HDIM];
    if (lane == 0) { sm[wave] = m; sl[wave] = l; }
    *(float4*)(&so[wave][lane * 4]) = acc;
    __syncthreads();

    if (wave == 0) {
        float gm = -INFINITY;
        #pragma unroll
        for (int w = 0; w < 8; ++w) gm = fmaxf(gm, sm[w]);
        float gl = 0.0f;
        float4 ga = {0.f, 0.f, 0.f, 0.f};
        #pragma unroll
        for (int w = 0; w < 8; ++w) {
            const float f = (sm[w] == -INFINITY) ? 0.0f : __expf(sm[w] - gm);
            gl += sl[w] * f;
            const float4 a = *(const float4*)(&so[w][lane * 4]);
            ga.x += f * a.x; ga.y += f * a.y; ga.z += f * a.z; ga.w += f * a.w;
        }
        const int pid = h * NSPLIT + split;
        if (lane == 0) { pm[pid] = gm; pl[pid] = gl; }
        *(float4*)(po + (size_t)pid * HDIM + lane * 4) = ga;
    }
}

__global__ void attn_combine_kernel(const float* __restrict__ pm, const float* __restrict__ pl,
                                    const float* __restrict__ po, float* __restrict__ attn) {
    const int h = blockIdx.x;
    const int d = threadIdx.x;       // 128 threads
    float gm = -INFINITY;
    for (int sp = 0; sp < NSPLIT; ++sp) {
        const int pid = h * NSPLIT + sp;
        if (pl[pid] > 0.0f) gm = fmaxf(gm, pm[pid]);
    }
    float gl = 0.0f, go = 0.0f;
    for (int sp = 0; sp < NSPLIT; ++sp) {
        const int pid = h * NSPLIT + sp;
        const float lc = pl[pid];
        if (lc > 0.0f) {
            const float f = __expf(pm[pid] - gm);
            gl += lc * f;
            go += f * po[(size_t)pid * HDIM + d];
        }
    }
    attn[h * HDIM + d] = go / gl;
}

__global__ void copyout_kernel(const float* __restrict__ x, float* __restrict__ out) {
    const int i = blockIdx.x * blockDim.x + threadIdx.x;
    if (i < DMODEL) out[i] = x[i];
}

// ---------------------------------------------------------------------------
extern "C" void kernel_launch(void* const* d_in, const int* in_sizes, int n_in,
                              void* d_out, int out_size, void* d_ws, size_t ws_size,
                              hipStream_t stream) {
    const int*   ids    = (const int*)  d_in[0];
    const int*   cpos   = (const int*)  d_in[1];
    const float* embed  = (const float*)d_in[2];
    const float* wq     = (const float*)d_in[3];   // (L, 2048, 2048)
    const float* wk     = (const float*)d_in[4];   // (L, 1024, 2048)
    const float* wv     = (const float*)d_in[5];   // (L, 1024, 2048)
    const float* wo     = (const float*)d_in[6];   // (L, 2048, 2048)
    const float* wg     = (const float*)d_in[7];   // (L, 5632, 2048)
    const float* wu     = (const float*)d_in[8];   // (L, 5632, 2048)
    const float* wd     = (const float*)d_in[9];   // (L, 2048, 5632)
    const float* ln1    = (const float*)d_in[10];  // (L, 2048)
    const float* ln2    = (const float*)d_in[11];  // (L, 2048)
    const float* kcache = (const float*)d_in[12];  // (L, 8, 8192, 128)
    const float* vcache = (const float*)d_in[13];  // (L, 8, 8192, 128)
    float* out = (float*)d_out;

    const int BIG = 1 << 28;   // "no boundary" sentinel for tile ranges

    // workspace layout (floats)
    float* ws  = (float*)d_ws;
    float* X   = ws;                       // 2048  residual stream
    float* QKV = X   + DMODEL;             // 4096  = Q(2048) | K(1024) | V(1024)
    float* Q   = QKV;
    float* Kb  = QKV + NHEADS * HDIM;
    float* Vb  = Kb  + NKVH * HDIM;
    float* ATT = QKV + 4096;               // 2048
    float* GU  = ATT + DMODEL;             // 11264 = G(5632) | U(5632)
    float* G   = GU;
    float* U   = GU + DFF;
    float* PM  = GU + 2 * DFF;             // 512
    float* PL  = PM + NHEADS * NSPLIT;     // 512
    float* PO  = PL + NHEADS * NSPLIT;     // 65536

    embed_kernel<<<1, 256, 0, stream>>>(embed, ids, X);

    for (int l = 0; l < LAYERS; ++l) {
        const float* wq_l = wq + (size_t)l * NHEADS * HDIM * DMODEL;
        const float* wk_l = wk + (size_t)l * NKVH  * HDIM * DMODEL;
        const float* wv_l = wv + (size_t)l * NKVH  * HDIM * DMODEL;
        const float* wo_l = wo + (size_t)l * DMODEL * (NHEADS * HDIM);
        const float* wg_l = wg + (size_t)l * DFF * DMODEL;
        const float* wu_l = wu + (size_t)l * DFF * DMODEL;
        const float* wd_l = wd + (size_t)l * DMODEL * DFF;
        const float* kc_l = kcache + (size_t)l * NKVH * SEQ * HDIM;
        const float* vc_l = vcache + (size_t)l * NKVH * SEQ * HDIM;

        // QKV projection, RMSNorm fused into B-operand. 256 tiles -> 128 blocks.
        gemv_fused_kernel<B_RMS><<<128, 256, 0, stream>>>(
            wq_l, wk_l, wv_l, /*b0=*/128, /*b1=*/192,
            X, ln1 + l * DMODEL, nullptr, QKV, DMODEL);

        // attention: flash-decode with fused RoPE, then split combine
        attn_partial_kernel<<<dim3(NHEADS, NSPLIT), 256, 0, stream>>>(
            Q, Kb, Vb, kc_l, vc_l, cpos, PM, PL, PO);
        attn_combine_kernel<<<NHEADS, HDIM, 0, stream>>>(PM, PL, PO, ATT);

        // output projection + residual. 128 tiles -> 64 blocks.
        gemv_fused_kernel<B_PLAIN><<<64, 256, 0, stream>>>(
            wo_l, nullptr, nullptr, BIG, BIG,
            ATT, nullptr, X, X, NHEADS * HDIM);

        // gate|up projection, RMSNorm fused. 704 tiles -> 352 blocks.
        gemv_fused_kernel<B_RMS><<<352, 256, 0, stream>>>(
            wg_l, wu_l, nullptr, /*b0=*/352, BIG,
            X, ln2 + l * DMODEL, nullptr, GU, DMODEL);

        // down projection, SwiGLU fused into B-operand, + residual. 64 blocks.
        gemv_fused_kernel<B_SWIGLU><<<64, 256, 0, stream>>>(
            wd_l, nullptr, nullptr, BIG, BIG,
            G, U, X, X, DFF);
    }

    copyout_kernel<<<(DMODEL + 255) / 256, 256, 0, stream>>>(X, out);
}